// CrystalGATEncoder_79637283602853
// MI455X (gfx1250) — compile-verified
//
#include <hip/hip_runtime.h>
#include <math.h>

#define NN 10000
#define NE 160000
#define H4 4
#define KC 128   // K-chunk staged into LDS for the B panel

typedef float v2f __attribute__((ext_vector_type(2)));
typedef float v8f __attribute__((ext_vector_type(8)));

static __device__ __forceinline__ void atomicMaxF(float* a, float v) {
  if (v >= 0.f) atomicMax((int*)a, __float_as_int(v));
  else          atomicMin((unsigned int*)a, __float_as_uint(v));
}

__global__ void fill_kernel(float* __restrict__ p, float v, long long n) {
  long long i = blockIdx.x * (long long)blockDim.x + threadIdx.x;
  if (i < n) p[i] = v;
}

// ---------------------------------------------------------------------------
// GEMM v1 (direct, no LDS): used only for tiny-N cases (layer-1 nproj, N=4).
// ---------------------------------------------------------------------------
__global__ __launch_bounds__(256) void wmma_gemm_f32(
    const float* __restrict__ A, int lda,
    const float* __restrict__ B, int ldb, int Kb,
    const float* __restrict__ bias,
    float* __restrict__ C, int ldc,
    int M, int N, int K, int act)
{
  int wave = (blockIdx.x * blockDim.x + threadIdx.x) >> 5;
  int lane = threadIdx.x & 31;
  int tiles_n = (N + 15) >> 4;
  int tiles_m = (M + 15) >> 4;
  if (wave >= tiles_m * tiles_n) return;   // wave-uniform: EXEC stays all-ones
  int mt = wave / tiles_n, nt = wave - mt * tiles_n;
  int m0 = mt << 4, n0 = nt << 4;
  int lhalf = lane >> 4;
  int l15   = lane & 15;
  int am = m0 + l15; if (am > M - 1) am = M - 1;
  int bn = n0 + l15; if (bn > N - 1) bn = N - 1;
  const float* Arow = A + (long long)am * lda;
  v8f acc = {};
  for (int kk = 0; kk < K; kk += 4) {
    int ka = kk + (lhalf << 1);
    v2f a, b;
    a.x = Arow[ka];
    a.y = Arow[ka + 1];
    int kb0 = ka;     if (kb0 > Kb - 1) kb0 = Kb - 1;
    int kb1 = ka + 1; if (kb1 > Kb - 1) kb1 = Kb - 1;
    b.x = B[(long long)kb0 * ldb + bn];
    b.y = B[(long long)kb1 * ldb + bn];
    __builtin_prefetch(Arow + ka + 64, 0, 1);
    acc = __builtin_amdgcn_wmma_f32_16x16x4_f32(
        false, a, false, b, (short)0, acc, false, false);
  }
  for (int r = 0; r < 8; ++r) {
    int m = m0 + r + (lhalf << 3);
    int n = n0 + l15;
    if (m < M && n < N) {
      float v = acc[r];
      if (bias) v += bias[n];
      if (act == 1) v = v > 0.f ? v : 0.f;
      C[(long long)m * ldc + n] = v;
    }
  }
}

// ---------------------------------------------------------------------------
// GEMM v2: block = 8 waves = 8 M-tiles x 1 N-tile. The 128x16 B panel chunk is
// staged into LDS with gfx1250 async-to-LDS b128 DMA (ASYNCcnt), then all 8
// waves consume it via ds loads while A streams from global per wave.
// Requires N % 16 == 0 and K % 4 == 0 (A zero-padded past Kb).
// ---------------------------------------------------------------------------
__global__ __launch_bounds__(256) void wmma_gemm_lds(
    const float* __restrict__ A, int lda,
    const float* __restrict__ B, int ldb, int Kb,
    const float* __restrict__ bias,
    float* __restrict__ C, int ldc,
    int M, int N, int K, int act, int tiles_n)
{
  __shared__ __align__(16) float bsm[KC * 16];
  int t = threadIdx.x;
  int wave = t >> 5, lane = t & 31;
  int nt = blockIdx.x % tiles_n;
  int ms = blockIdx.x / tiles_n;
  int mt = ms * 8 + wave;
  int tiles_m = (M + 15) >> 4;
  int m0 = mt << 4, n0 = nt << 4;
  int lhalf = lane >> 4, l15 = lane & 15;
  int am = m0 + l15; if (am > M - 1) am = M - 1;   // tail waves compute garbage, masked on store
  const float* Arow = A + (long long)am * lda;
  v8f acc = {};
  for (int kc = 0; kc < K; kc += KC) {
    // --- stage B[kc:kc+KC, n0:n0+16] -> LDS via async b128 DMA (2 per thread)
    #pragma unroll
    for (int i = 0; i < 2; ++i) {
      int idx = t + i * 256;                 // 0..511 16-byte segments
      int row = idx >> 2;                    // 0..127
      int seg = idx & 3;                     // 4 floats each
      int gr = kc + row; if (gr > Kb - 1) gr = Kb - 1;   // finite pad rows
      unsigned long long gaddr =
          (unsigned long long)(const void*)(B + (long long)gr * ldb + n0 + seg * 4);
      unsigned laddr = (unsigned)(uintptr_t)&bsm[idx * 4];
      asm volatile("global_load_async_to_lds_b128 %0, %1, off"
                   :: "v"(laddr), "v"(gaddr) : "memory");
    }
    asm volatile("s_wait_asynccnt 0x0" ::: "memory");
    __syncthreads();
    // --- consume chunk
    int kend = (kc + KC < K) ? kc + KC : K;
    for (int kk = kc; kk < kend; kk += 4) {
      int ka = kk + (lhalf << 1);
      v2f a, b;
      a.x = Arow[ka];
      a.y = Arow[ka + 1];
      int lr = ka - kc;
      b.x = bsm[lr * 16 + l15];
      b.y = bsm[(lr + 1) * 16 + l15];
      __builtin_prefetch(Arow + ka + 64, 0, 1);
      acc = __builtin_amdgcn_wmma_f32_16x16x4_f32(
          false, a, false, b, (short)0, acc, false, false);
    }
    __syncthreads();   // protect bsm before next chunk overwrites it
  }
  if (mt < tiles_m) {
    for (int r = 0; r < 8; ++r) {
      int m = m0 + r + (lhalf << 3);
      int n = n0 + l15;
      if (m < M && n < N) {
        float v = acc[r];
        if (bias) v += bias[n];
        if (act == 1) v = v > 0.f ? v : 0.f;
        C[(long long)m * ldc + n] = v;
      }
    }
  }
}

// edge MLP: 3 -> relu(6) -> 3
__global__ void edge_mlp_kernel(const float* __restrict__ raw,
    const float* __restrict__ w1, const float* __restrict__ b1,
    const float* __restrict__ w2, const float* __restrict__ b2,
    float* __restrict__ ef, int E)
{
  int e = blockIdx.x * blockDim.x + threadIdx.x;
  if (e >= E) return;
  float x0 = raw[e*3], x1 = raw[e*3+1], x2 = raw[e*3+2];
  float h[6];
  #pragma unroll
  for (int j = 0; j < 6; ++j) {
    float s = b1[j] + x0*w1[j] + x1*w1[6+j] + x2*w1[12+j];
    h[j] = s > 0.f ? s : 0.f;
  }
  #pragma unroll
  for (int c = 0; c < 3; ++c) {
    float s = b2[c];
    #pragma unroll
    for (int j = 0; j < 6; ++j) s += h[j]*w2[j*3+c];
    ef[e*3+c] = s;
  }
}

__global__ void deg_kernel(const int* __restrict__ src, const int* __restrict__ dst,
                           float* __restrict__ deg, int E) {
  int e = blockIdx.x * blockDim.x + threadIdx.x;
  if (e >= E) return;
  atomicAdd(&deg[src[e]], 1.f);
  atomicAdd(&deg[dst[e]], 1.f);
}

__global__ void scatter_ef_kernel(const float* __restrict__ ef,
    const int* __restrict__ src, const int* __restrict__ dst,
    float* __restrict__ temp, int Kpad, int I, int E)
{
  int idx = blockIdx.x * blockDim.x + threadIdx.x;
  if (idx >= E * 3) return;
  int e = idx / 3, c = idx - e * 3;
  float v = ef[idx];
  atomicAdd(&temp[(long long)src[e]*Kpad + I + c], v);
  atomicAdd(&temp[(long long)dst[e]*Kpad + I + c], v);
}

// temp[v, 0:I] = deg[v] * nproj[v, :]
__global__ void temp_node_kernel(const float* __restrict__ nproj,
    const float* __restrict__ deg, float* __restrict__ temp,
    int I, int Kpad, long long n)
{
  long long idx = blockIdx.x * (long long)blockDim.x + threadIdx.x;
  if (idx >= n) return;
  int v = (int)(idx / I), i = (int)(idx - (long long)v * I);
  temp[(long long)v * Kpad + i] = deg[v] * nproj[idx];
}

__global__ void el_er_kernel(const float* __restrict__ feat,
    const float* __restrict__ al, const float* __restrict__ ar,
    float* __restrict__ el, float* __restrict__ er, int n, int O)
{
  int idx = blockIdx.x * blockDim.x + threadIdx.x;  // idx = v*4 + h
  if (idx >= n) return;
  int h = idx & 3;
  const float* f = feat + (long long)idx * O;       // (v*4+h)*O == v*F + h*O
  const float* a = al + h * O;
  const float* b = ar + h * O;
  float sl = 0.f, sr = 0.f;
  for (int o = 0; o < O; ++o) { float x = f[o]; sl += x*a[o]; sr += x*b[o]; }
  el[idx] = sl; er[idx] = sr;
}

__global__ void edge_e_kernel(const float* __restrict__ el, const float* __restrict__ er,
    const int* __restrict__ src, const int* __restrict__ dst,
    float* __restrict__ alpha, float* __restrict__ mx, int n)
{
  int idx = blockIdx.x * blockDim.x + threadIdx.x;  // e*4 + h
  if (idx >= n) return;
  int e = idx >> 2, h = idx & 3;
  float v = el[src[e]*4 + h] + er[dst[e]*4 + h];
  v = v > 0.f ? v : 0.2f * v;                       // leaky_relu 0.2
  alpha[idx] = v;
  atomicMaxF(&mx[dst[e]*4 + h], v);
}

__global__ void edge_w_kernel(float* __restrict__ alpha,
    const float* __restrict__ mx, float* __restrict__ sm,
    const int* __restrict__ dst, int n)
{
  int idx = blockIdx.x * blockDim.x + threadIdx.x;
  if (idx >= n) return;
  int e = idx >> 2, h = idx & 3;
  float w = expf(alpha[idx] - mx[dst[e]*4 + h]);
  alpha[idx] = w;
  atomicAdd(&sm[dst[e]*4 + h], w);
}

__global__ void alpha_div_kernel(float* __restrict__ alpha,
    const float* __restrict__ sm, const int* __restrict__ dst, int n)
{
  int idx = blockIdx.x * blockDim.x + threadIdx.x;
  if (idx >= n) return;
  int e = idx >> 2, h = idx & 3;
  alpha[idx] = alpha[idx] / sm[dst[e]*4 + h];
}

__global__ void aggregate_kernel(const float* __restrict__ feat,
    const float* __restrict__ alpha,
    const int* __restrict__ src, const int* __restrict__ dst,
    float* __restrict__ out, int lf, int lo, long long total)
{
  long long idx = blockIdx.x * (long long)blockDim.x + threadIdx.x;
  if (idx >= total) return;
  int e = (int)(idx >> lf);
  int f = (int)(idx & ((1 << lf) - 1));
  int h = f >> lo;
  float a = alpha[(e << 2) + h];
  atomicAdd(&out[((long long)dst[e] << lf) + f],
            a * feat[((long long)src[e] << lf) + f]);
}

__global__ void bias_elu_kernel(float* __restrict__ h, const float* __restrict__ bias,
                                long long n, int F)
{
  long long idx = blockIdx.x * (long long)blockDim.x + threadIdx.x;
  if (idx >= n) return;
  int f = (int)(idx % F);
  float v = h[idx] + bias[f];
  h[idx] = v > 0.f ? v : expf(v) - 1.f;   // elu
}

__global__ void gate_kernel(const float* __restrict__ gh, const float* __restrict__ w,
                            const float* __restrict__ b, float* __restrict__ gate, int n)
{
  int v = blockIdx.x * blockDim.x + threadIdx.x;
  if (v >= n) return;
  const float* g = gh + (long long)v * 128;
  float s = b[0];
  for (int j = 0; j < 128; ++j) s += g[j] * w[j];
  gate[v] = s;
}

__global__ void gate_reduce_kernel(const float* __restrict__ gate,
                                   float* __restrict__ scal, int n)
{
  __shared__ float sh[1024];
  int t = threadIdx.x;
  float mx = -INFINITY;
  for (int i = t; i < n; i += 1024) mx = fmaxf(mx, gate[i]);
  sh[t] = mx; __syncthreads();
  for (int s = 512; s > 0; s >>= 1) { if (t < s) sh[t] = fmaxf(sh[t], sh[t+s]); __syncthreads(); }
  float gmax = sh[0]; __syncthreads();
  float sum = 0.f;
  for (int i = t; i < n; i += 1024) sum += expf(gate[i] - gmax);
  sh[t] = sum; __syncthreads();
  for (int s = 512; s > 0; s >>= 1) { if (t < s) sh[t] += sh[t+s]; __syncthreads(); }
  if (t == 0) { scal[0] = gmax; scal[1] = sh[0]; }
}

__global__ void alpha_node_kernel(float* __restrict__ gate,
                                  const float* __restrict__ scal, int n)
{
  int v = blockIdx.x * blockDim.x + threadIdx.x;
  if (v >= n) return;
  gate[v] = expf(gate[v] - scal[0]) / scal[1];
}

__global__ void pool_kernel(const float* __restrict__ h3, const float* __restrict__ a,
                            float* __restrict__ ge, int n)
{
  int jg = blockIdx.x & 3;
  int chunk = blockIdx.x >> 2;
  int nchunks = gridDim.x >> 2;
  int j = jg * 256 + threadIdx.x;
  float acc = 0.f;
  for (int v = chunk; v < n; v += nchunks)
    acc += a[v] * h3[(long long)v * 1024 + j];
  atomicAdd(&ge[j], acc);
}

__global__ void lat_kernel(const float* __restrict__ lattice, const float* __restrict__ lw,
                           const float* __restrict__ lb, float* __restrict__ ge_lat)
{
  int j = threadIdx.x;  // 128
  float s = lb[j];
  #pragma unroll
  for (int i = 0; i < 6; ++i) s += lattice[i] * lw[i*128 + j];
  ge_lat[j] = s;
}

__global__ void head_kernel(const float* __restrict__ ge,
    const float* __restrict__ mw, const float* __restrict__ mb,
    const float* __restrict__ vw, const float* __restrict__ vb,
    float* __restrict__ out)
{
  int j = threadIdx.x;  // 128
  if (j < 64) {
    float s = mb[j];
    for (int i = 0; i < 1152; ++i) s += ge[i] * mw[i*64 + j];
    out[j] = s;
  } else {
    int c = j - 64;
    float s = vb[c];
    for (int i = 0; i < 1152; ++i) s += ge[i] * vw[i*64 + c];
    out[64 + c] = fminf(10.f, fmaxf(-10.f, s));
  }
}

static inline int cdiv(long long a, int b) { return (int)((a + b - 1) / b); }

extern "C" void kernel_launch(void* const* d_in, const int* in_sizes, int n_in,
                              void* d_out, int out_size, void* d_ws, size_t ws_size,
                              hipStream_t stream) {
  (void)in_sizes; (void)n_in; (void)out_size; (void)ws_size;
  // Input layout: depth-first insertion order of setup_inputs()
  // 0 node_feats, 1 edge_feats, 2 lattice,
  // g1: 3 ew1, 4 eb1, 5 ew2, 6 eb2, 7 pw, 8 pb, 9 fc, 10 al, 11 ar, 12 bias
  // g2: 13..22, g3: 23..32,
  // 33 gw1, 34 gb1, 35 gw2, 36 gb2, 37 lw, 38 lb, 39 mw, 40 mb, 41 vw, 42 vb,
  // 43 edge_src, 44 edge_dst
  const float* node_feats = (const float*)d_in[0];
  const float* edge_feats = (const float*)d_in[1];
  const float* lattice    = (const float*)d_in[2];
  const float* gw1 = (const float*)d_in[33];
  const float* gb1 = (const float*)d_in[34];
  const float* gw2 = (const float*)d_in[35];
  const float* gb2 = (const float*)d_in[36];
  const float* lw  = (const float*)d_in[37];
  const float* lb  = (const float*)d_in[38];
  const float* mw  = (const float*)d_in[39];
  const float* mb  = (const float*)d_in[40];
  const float* vw  = (const float*)d_in[41];
  const float* vb  = (const float*)d_in[42];
  const int* src = (const int*)d_in[43];
  const int* dst = (const int*)d_in[44];
  float* out = (float*)d_out;

  float* ws    = (float*)d_ws;
  float* EF    = ws;                    // 480000
  float* NPROJ = EF    + 480000;        // 5,120,000
  float* TEMP  = NPROJ + 5120000;       // 5,160,000
  float* FEAT  = TEMP  + 5160000;       // 10,240,000
  float* HB1   = FEAT  + 10240000;      // 2,560,000
  float* HB2   = HB1   + 2560000;       // 5,120,000
  float* EL    = HB2   + 5120000;       // 40000
  float* ER    = EL    + 40000;
  float* MX    = ER    + 40000;
  float* SM    = MX    + 40000;
  float* ALPHA = SM    + 40000;         // 640000
  float* DEG   = ALPHA + 640000;        // 10000
  float* GATEB = DEG   + 10000;         // 10000
  float* GE    = GATEB + 10000;         // 1152
  float* SCAL  = GE    + 1152;          // 2
  float* GH    = TEMP;                  // alias (TEMP dead after layer-3 fc)
  float* H3    = out + 128;             // node_emb region of d_out

  auto gemm = [&](const float* A, int lda, const float* B, int ldb, int Kb,
                  const float* bias, float* C, int ldc, int M, int Nn, int K, int act) {
    int tiles_n = (Nn + 15) / 16, tiles_m = (M + 15) / 16;
    if ((Nn & 15) == 0) {
      int blocks = tiles_n * ((tiles_m + 7) / 8);
      wmma_gemm_lds<<<blocks, 256, 0, stream>>>(A, lda, B, ldb, Kb, bias,
                                                C, ldc, M, Nn, K, act, tiles_n);
    } else {
      int tiles = tiles_m * tiles_n;
      wmma_gemm_f32<<<(tiles + 7) / 8, 256, 0, stream>>>(A, lda, B, ldb, Kb, bias,
                                                         C, ldc, M, Nn, K, act);
    }
  };

  // degree (reused by all layers)
  fill_kernel<<<cdiv(NN, 256), 256, 0, stream>>>(DEG, 0.f, NN);
  deg_kernel<<<cdiv(NE, 256), 256, 0, stream>>>(src, dst, DEG, NE);

  auto layer = [&](const float* hprev, int I, int O, int pb, float* hout) {
    const float* ew1  = (const float*)d_in[pb+0];
    const float* eb1  = (const float*)d_in[pb+1];
    const float* ew2  = (const float*)d_in[pb+2];
    const float* eb2  = (const float*)d_in[pb+3];
    const float* pw   = (const float*)d_in[pb+4];
    const float* pbw  = (const float*)d_in[pb+5];
    const float* fc   = (const float*)d_in[pb+6];
    const float* al   = (const float*)d_in[pb+7];
    const float* ar   = (const float*)d_in[pb+8];
    const float* bias = (const float*)d_in[pb+9];
    int F = H4 * O;
    int Kpad = ((I + 3) + 3) & ~3;
    int lf = (F == 256) ? 8 : (F == 512) ? 9 : 10;
    int lo = (O == 64) ? 6 : (O == 128) ? 7 : 8;

    edge_mlp_kernel<<<cdiv(NE, 256), 256, 0, stream>>>(edge_feats, ew1, eb1, ew2, eb2, EF, NE);
    // nproj = hprev @ pw + pb
    gemm(hprev, I, pw, I, I, pbw, NPROJ, I, NN, I, I, 0);
    // temp = [deg*nproj | scatter(ef) | zero pad]
    fill_kernel<<<cdiv((long long)NN * Kpad, 256), 256, 0, stream>>>(TEMP, 0.f, (long long)NN * Kpad);
    scatter_ef_kernel<<<cdiv((long long)NE * 3, 256), 256, 0, stream>>>(EF, src, dst, TEMP, Kpad, I, NE);
    temp_node_kernel<<<cdiv((long long)NN * I, 256), 256, 0, stream>>>(NPROJ, DEG, TEMP, I, Kpad, (long long)NN * I);
    // feat = temp @ fc
    gemm(TEMP, Kpad, fc, F, I + 3, nullptr, FEAT, F, NN, F, Kpad, 0);
    // attention logits
    el_er_kernel<<<cdiv(NN * 4, 256), 256, 0, stream>>>(FEAT, al, ar, EL, ER, NN * 4, O);
    fill_kernel<<<cdiv(NN * 4, 256), 256, 0, stream>>>(MX, -INFINITY, NN * 4);
    fill_kernel<<<cdiv(NN * 4, 256), 256, 0, stream>>>(SM, 0.f, NN * 4);
    edge_e_kernel<<<cdiv(NE * 4, 256), 256, 0, stream>>>(EL, ER, src, dst, ALPHA, MX, NE * 4);
    edge_w_kernel<<<cdiv(NE * 4, 256), 256, 0, stream>>>(ALPHA, MX, SM, dst, NE * 4);
    alpha_div_kernel<<<cdiv(NE * 4, 256), 256, 0, stream>>>(ALPHA, SM, dst, NE * 4);
    // aggregate + bias + elu
    fill_kernel<<<cdiv((long long)NN * F, 256), 256, 0, stream>>>(hout, 0.f, (long long)NN * F);
    aggregate_kernel<<<cdiv((long long)NE * F, 256), 256, 0, stream>>>(FEAT, ALPHA, src, dst, hout, lf, lo, (long long)NE * F);
    bias_elu_kernel<<<cdiv((long long)NN * F, 256), 256, 0, stream>>>(hout, bias, (long long)NN * F, F);
  };

  layer(node_feats, 4,   64,  3,  HB1);   // g1: 4   -> 256
  layer(HB1,        256, 128, 13, HB2);   // g2: 256 -> 512
  layer(HB2,        512, 256, 23, H3);    // g3: 512 -> 1024 (written into d_out)

  // gate = relu(h3 @ gw1 + gb1) @ gw2 + gb2 ; softmax pool
  gemm(H3, 1024, gw1, 128, 1024, gb1, GH, 128, NN, 128, 1024, 1);
  gate_kernel<<<cdiv(NN, 256), 256, 0, stream>>>(GH, gw2, gb2, GATEB, NN);
  gate_reduce_kernel<<<1, 1024, 0, stream>>>(GATEB, SCAL, NN);
  alpha_node_kernel<<<cdiv(NN, 256), 256, 0, stream>>>(GATEB, SCAL, NN);
  fill_kernel<<<cdiv(1024, 256), 256, 0, stream>>>(GE, 0.f, 1024);
  pool_kernel<<<256, 256, 0, stream>>>(H3, GATEB, GE, NN);
  lat_kernel<<<1, 128, 0, stream>>>(lattice, lw, lb, GE + 1024);
  head_kernel<<<1, 128, 0, stream>>>(GE, mw, mb, vw, vb, out);
}